// VectorQuantize_44452911514211
// MI455X (gfx1250) — compile-verified
//
#include <hip/hip_runtime.h>
#include <hip/hip_bf16.h>
#include <math.h>

// Problem constants (from reference setup_inputs)
#define B_SZ   4
#define CIN    768
#define LIN    4096
#define DCODE  128
#define TOUT   2048          // (4096 + 2*1 - 3)/2 + 1
#define NROWS  (B_SZ * TOUT) // 8192 encoding rows
#define KENT   8192          // codebook entries
#define EPSN   1e-12f

typedef __attribute__((ext_vector_type(2))) float v2f;
typedef __attribute__((ext_vector_type(8))) float v8f;

// ---------------------------------------------------------------------------
// Kernel 0: transpose conv weights  w[o][i][k] -> wT[(i*3+k)][o]  (coalesced
// per-channel loads in the conv kernel).
// ---------------------------------------------------------------------------
__global__ void k_transpose_w(const float* __restrict__ w, float* __restrict__ wT) {
    int idx = blockIdx.x * blockDim.x + threadIdx.x;       // over 128*768*3
    const int total = DCODE * CIN * 3;
    if (idx >= total) return;
    int o  = idx / (CIN * 3);
    int ik = idx - o * (CIN * 3);                          // i*3 + k
    wT[ik * DCODE + o] = w[idx];
}

// ---------------------------------------------------------------------------
// Kernel 1: L2-normalize codebook rows. One block (128 threads) per row.
// ---------------------------------------------------------------------------
__global__ void k_norm_codebook(const float* __restrict__ cb, float* __restrict__ cbn) {
    __shared__ float red[DCODE];
    int r = blockIdx.x;
    int o = threadIdx.x;
    float v = cb[(size_t)r * DCODE + o];
    red[o] = v * v;
    __syncthreads();
    for (int s = DCODE / 2; s > 0; s >>= 1) {
        if (o < s) red[o] += red[o + s];
        __syncthreads();
    }
    __shared__ float invn;
    if (o == 0) invn = 1.0f / fmaxf(sqrtf(red[0]), EPSN);
    __syncthreads();
    cbn[(size_t)r * DCODE + o] = v * invn;
}

// ---------------------------------------------------------------------------
// Kernel 2: Conv1d(768->128, k=3, stride=2, pad=1) + per-row L2 normalize.
// Grid: 512 blocks of 128 threads. Block g covers batch b = g>>7 and 16 time
// positions t0..t0+15. Thread o owns output channel o for all 16 positions.
// x window for an input-channel chunk is staged in LDS (broadcast reads).
// ---------------------------------------------------------------------------
#define CI_CHUNK 32
__global__ void k_conv_norm(const float* __restrict__ x,
                            const float* __restrict__ wT,
                            float* __restrict__ enc) {
    __shared__ float xs[CI_CHUNK][33];        // window: positions 2*t0-1 .. 2*t0+31
    __shared__ float outv[16][DCODE];
    __shared__ float invn[16];

    const int g  = blockIdx.x;
    const int b  = g >> 7;                    // 128 blocks per batch
    const int t0 = (g & 127) << 4;
    const int o  = threadIdx.x;
    const int base = 2 * t0 - 1;              // leftmost x position needed

    float acc[16];
#pragma unroll
    for (int m = 0; m < 16; ++m) acc[m] = 0.0f;

    for (int ic0 = 0; ic0 < CIN; ic0 += CI_CHUNK) {
        __syncthreads();                      // protect xs reuse
        for (int idx = o; idx < CI_CHUNK * 33; idx += DCODE) {
            int ii  = idx / 33;
            int tt  = idx - ii * 33;
            int pos = base + tt;
            float xv = 0.0f;
            if (pos >= 0 && pos < LIN)
                xv = x[((size_t)b * CIN + (ic0 + ii)) * LIN + pos];
            xs[ii][tt] = xv;
        }
        __syncthreads();

#pragma unroll 4
        for (int ii = 0; ii < CI_CHUNK; ++ii) {
            int i = ic0 + ii;
            float w0 = wT[(size_t)(i * 3 + 0) * DCODE + o];
            float w1 = wT[(size_t)(i * 3 + 1) * DCODE + o];
            float w2 = wT[(size_t)(i * 3 + 2) * DCODE + o];
#pragma unroll
            for (int m = 0; m < 16; ++m) {
                acc[m] = fmaf(xs[ii][2 * m + 0], w0, acc[m]);
                acc[m] = fmaf(xs[ii][2 * m + 1], w1, acc[m]);
                acc[m] = fmaf(xs[ii][2 * m + 2], w2, acc[m]);
            }
        }
    }

#pragma unroll
    for (int m = 0; m < 16; ++m) outv[m][o] = acc[m];
    __syncthreads();
    if (o < 16) {
        float s = 0.0f;
        for (int c = 0; c < DCODE; ++c) { float v = outv[o][c]; s = fmaf(v, v, s); }
        invn[o] = 1.0f / fmaxf(sqrtf(s), EPSN);
    }
    __syncthreads();
#pragma unroll
    for (int m = 0; m < 16; ++m) {
        size_t row = (size_t)b * TOUT + (t0 + m);
        enc[row * DCODE + o] = outv[m][o] * invn[m];
    }
}

// ---------------------------------------------------------------------------
// Kernel 3: cosine-similarity argmax via V_WMMA_F32_16X16X4_F32.
// Block = 256 threads = 8 waves; block owns a 16-row M-tile of enc.
// Each wave scans 1024 codebook entries (64 N-tiles of 16), accumulating
// K=128 with 32 wmma ops per N-tile, then keeps a per-lane running argmax
// over its 8 C-registers. Final reduction across lanes+waves via LDS.
//
// All 32 B fragments for a tile are loaded into independent registers
// (load clause + pipelined s_wait_loadcnt) so the WMMAs issue back-to-back;
// next tile's loads overlap the current WMMA chain. Next tile's B rows are
// prefetched at WGP scope via inline asm (scope 0 pulls into all cache
// levels, so the scattered 16 B fragment loads hit the WGP cache).
//
// f32 16x16x4 fragment layouts (ISA 7.12.2):
//   A (16x4): lane&15 = M row; VGPR0 holds K = 2*(lane>>4), VGPR1 = +1
//   B (4x16): lane&15 = N col; same K selection per lane half
//   C (16x16): VGPR j, lane l -> M = j + 8*(l>>4), N = l&15
// ---------------------------------------------------------------------------
__global__ void k_argmax_wmma(const float* __restrict__ enc,
                              const float* __restrict__ cbn,
                              int* __restrict__ out) {
    __shared__ float rv[8][16][16];
    __shared__ int   ri[8][16][16];

    const int tid  = threadIdx.x;
    const int wav  = tid >> 5;
    const int lane = tid & 31;
    const int mrow = lane & 15;
    const int koff = (lane >> 4) * 2;        // 0 or 2
    const int r0   = blockIdx.x * 16;        // M-tile base row

    // Preload all A fragments for this M-tile (K = 0..127 in steps of 4).
    const float* arow = enc + (size_t)(r0 + mrow) * DCODE + koff;
    v2f a[32];
#pragma unroll
    for (int s = 0; s < 32; ++s)
        a[s] = *(const v2f*)(arow + s * 4);

    float bv[8];
    int   bi[8];
#pragma unroll
    for (int j = 0; j < 8; ++j) { bv[j] = -__builtin_huge_valf(); bi[j] = 0; }

    const int nbase = wav * (KENT / 8);      // 1024 entries per wave
    const int NT    = (KENT / 8) / 16;       // 64 N-tiles per wave

    for (int nt = 0; nt < NT; ++nt) {
        const int n0 = nbase + nt * 16;
        const float* brow = cbn + (size_t)(n0 + mrow) * DCODE + koff;

        // WGP-scope (scope 0) prefetch of next tile's rows: each lane
        // prefetches its own row at +0 and +256 B (row spans 512 B).
        // Addresses stay inside cbn (non-speculative-safe).
        if (nt + 1 < NT) {
            const float* nrow = brow + (size_t)16 * DCODE;
            asm volatile(
                "global_prefetch_b8 %0, off\n\t"
                "global_prefetch_b8 %0, off offset:256"
                :: "v"(nrow));
        }

        // (a) Batch-load all 32 B fragments into independent registers.
        v2f bf[32];
#pragma unroll
        for (int s = 0; s < 32; ++s)
            bf[s] = *(const v2f*)(brow + s * 4);

        // (b) 32 back-to-back WMMAs (waits pipeline against the load clause).
        v8f c = {};
#pragma unroll
        for (int s = 0; s < 32; ++s)
            c = __builtin_amdgcn_wmma_f32_16x16x4_f32(
                    false, a[s], false, bf[s], (short)0, c, false, false);

        const int col = n0 + mrow;           // this lane's codebook entry
#pragma unroll
        for (int j = 0; j < 8; ++j) {
            if (c[j] > bv[j]) { bv[j] = c[j]; bi[j] = col; }
        }
    }

    // Stash per-lane bests: row = j + 8*(lane>>4), col-slot = lane&15.
#pragma unroll
    for (int j = 0; j < 8; ++j) {
        int row = j + 8 * (lane >> 4);
        rv[wav][row][mrow] = bv[j];
        ri[wav][row][mrow] = bi[j];
    }
    __syncthreads();

    if (tid < 16) {
        float best = -__builtin_huge_valf();
        int   besti = 0;
        for (int w2 = 0; w2 < 8; ++w2)
            for (int cIdx = 0; cIdx < 16; ++cIdx) {
                float v = rv[w2][tid][cIdx];
                if (v > best) { best = v; besti = ri[w2][tid][cIdx]; }
            }
        out[r0 + tid] = besti;
    }
}

// ---------------------------------------------------------------------------
// Host-side launcher
// ---------------------------------------------------------------------------
extern "C" void kernel_launch(void* const* d_in, const int* in_sizes, int n_in,
                              void* d_out, int out_size, void* d_ws, size_t ws_size,
                              hipStream_t stream) {
    const float* x  = (const float*)d_in[0];   // [4, 768, 4096]
    const float* w  = (const float*)d_in[1];   // [128, 768, 3]
    const float* cb = (const float*)d_in[2];   // [8192, 128]
    int* out = (int*)d_out;                    // [4, 2048] int32 indices

    // Workspace partition (floats): wT | enc | cbn  (~9.6 MB total)
    float* wT  = (float*)d_ws;
    float* enc = wT + (size_t)CIN * 3 * DCODE;          // 294912
    float* cbn = enc + (size_t)NROWS * DCODE;           // +1048576

    {   // 0: weight transpose
        int total = DCODE * CIN * 3;
        k_transpose_w<<<(total + 255) / 256, 256, 0, stream>>>(w, wT);
    }
    // 1: codebook normalize
    k_norm_codebook<<<KENT, DCODE, 0, stream>>>(cb, cbn);
    // 2: conv + encoding normalize (512 blocks x 128 threads)
    k_conv_norm<<<NROWS / 16, DCODE, 0, stream>>>(x, wT, enc);
    // 3: WMMA similarity + argmax (512 blocks x 256 threads)
    k_argmax_wmma<<<NROWS / 16, 256, 0, stream>>>(enc, cbn, out);
}